// one_layer_gcn_63969242906880
// MI455X (gfx1250) — compile-verified
//
#include <hip/hip_runtime.h>

typedef __attribute__((ext_vector_type(2))) float v2f;
typedef __attribute__((ext_vector_type(8))) float v8f;

#define DIM     512
#define TILE_K  64
#define ROWPAD  68   // 16x64 tile rows padded to 68 floats -> bank-conflict-free A reads

// ---------------- init: deg = 1 (self loop), out = 0 ----------------
__global__ void gcn_init_kernel(float* __restrict__ deg, float* __restrict__ out, int n) {
  int i = blockIdx.x * blockDim.x + threadIdx.x;
  if (i < n) { deg[i] = 1.0f; out[i] = 0.0f; }
}

// ---------------- h = x @ W via V_WMMA_F32_16X16X4_F32 ----------------
// 256 threads = 8 waves; each wave owns 16 rows, loops K in 64-wide slices.
__global__ void __launch_bounds__(256)
gcn_gemv_wmma_kernel(const float* __restrict__ x, const float* __restrict__ W,
                     float* __restrict__ h, int n) {
  __shared__ float Ws[DIM];
  __shared__ float tile[8 * 16 * ROWPAD];

  const int tid = threadIdx.x;
  Ws[tid]       = W[tid];
  Ws[tid + 256] = W[tid + 256];
  __syncthreads();

  const int wave = tid >> 5;
  const int lane = tid & 31;
  const int half = lane >> 4;   // 0: K=0,1   1: K=2,3  (ISA 16x4 f32 A layout)
  const int m    = lane & 15;
  float* wt = &tile[wave * 16 * ROWPAD];

  const long rowBase = (long)blockIdx.x * 128 + (long)wave * 16;

  v8f c = {};

  for (int k0 = 0; k0 < DIM; k0 += TILE_K) {
    // stage 16 rows x 64 cols into LDS, coalesced float4 loads
    #pragma unroll
    for (int t = 0; t < 8; ++t) {
      int idx4 = t * 32 + lane;          // 0..255 float4 slots
      int row  = idx4 >> 4;              // 0..15
      int c4   = idx4 & 15;              // 0..15
      long g = rowBase + row;
      if (g >= n) g = n - 1;             // clamp: EXEC must stay all-ones for WMMA
      float4 v = *(const float4*)(x + g * DIM + k0 + c4 * 4);
      *(float4*)(&wt[row * ROWPAD + c4 * 4]) = v;
    }
    // 16 WMMAs over this K slice
    #pragma unroll
    for (int kk = 0; kk < TILE_K; kk += 4) {
      int k = kk + half * 2;
      float2 av = *(const float2*)(&wt[m * ROWPAD + k]);
      float2 wv = *(const float2*)(&Ws[k0 + k]);
      v2f a;  a.x = av.x; a.y = av.y;
      v2f bv; bv.x = wv.x; bv.y = wv.y;  // broadcast W across all 16 N-columns
      c = __builtin_amdgcn_wmma_f32_16x16x4_f32(false, a, false, bv,
                                                (short)0, c, false, false);
    }
  }

  // D column 0: lane 0 holds rows M=0..7, lane 16 holds rows M=8..15
  if (m == 0) {
    long base = rowBase + (long)half * 8;
    #pragma unroll
    for (int r = 0; r < 8; ++r) {
      long g = base + r;
      if (g < n) h[g] = c[r];
    }
  }
}

// ---------------- degree: L2-resident float atomics ----------------
__global__ void gcn_degree_kernel(const long long* __restrict__ dst,
                                  float* __restrict__ deg, int e) {
  int i = blockIdx.x * blockDim.x + threadIdx.x;
  if (i < e) {
    unsafeAtomicAdd(&deg[(int)dst[i]], 1.0f);
  }
}

// ---------------- dinv = rsqrt(deg) in place (deg >= 1 always) ----------------
__global__ void gcn_rsqrt_kernel(float* __restrict__ deg, int n) {
  int i = blockIdx.x * blockDim.x + threadIdx.x;
  if (i < n) deg[i] = rsqrtf(deg[i]);
}

// ---------------- scatter: out[dst] += h[src]*dinv[src]*dinv[dst] ----------------
__global__ void gcn_scatter_kernel(const long long* __restrict__ src,
                                   const long long* __restrict__ dst,
                                   const float* __restrict__ h,
                                   const float* __restrict__ dinv,
                                   float* __restrict__ out, int e) {
  int i = blockIdx.x * blockDim.x + threadIdx.x;
  if (i < e) {
    int s = (int)src[i];
    int d = (int)dst[i];
    float v = h[s] * dinv[s] * dinv[d];
    unsafeAtomicAdd(&out[d], v);
  }
}

// ---------------- finalize: add self-loop term + bias, ReLU ----------------
__global__ void gcn_finalize_kernel(float* __restrict__ out,
                                    const float* __restrict__ h,
                                    const float* __restrict__ dinv,
                                    const float* __restrict__ b, int n) {
  int i = blockIdx.x * blockDim.x + threadIdx.x;
  if (i < n) {
    float di = dinv[i];
    float v = out[i] + h[i] * di * di + b[0];
    out[i] = v > 0.0f ? v : 0.0f;
  }
}

extern "C" void kernel_launch(void* const* d_in, const int* in_sizes, int n_in,
                              void* d_out, int out_size, void* d_ws, size_t ws_size,
                              hipStream_t stream) {
  const float*     x  = (const float*)d_in[0];
  const long long* ei = (const long long*)d_in[1];  // int64 [2, E] flat
  const float*     W  = (const float*)d_in[2];
  const float*     b  = (const float*)d_in[3];
  float* out = (float*)d_out;

  const int n = in_sizes[0] / DIM;   // 200000
  const int e = in_sizes[1] / 2;     // 12.8M

  float* h   = (float*)d_ws;         // [n]
  float* deg = h + n;                // [n], becomes dinv in place

  const long long* src = ei;
  const long long* dst = ei + e;

  gcn_init_kernel     <<<(n + 255) / 256, 256, 0, stream>>>(deg, out, n);
  gcn_gemv_wmma_kernel<<<(n + 127) / 128, 256, 0, stream>>>(x, W, h, n);
  gcn_degree_kernel   <<<(e + 255) / 256, 256, 0, stream>>>(dst, deg, e);
  gcn_rsqrt_kernel    <<<(n + 255) / 256, 256, 0, stream>>>(deg, n);
  gcn_scatter_kernel  <<<(e + 255) / 256, 256, 0, stream>>>(src, dst, h, deg, out, e);
  gcn_finalize_kernel <<<(n + 255) / 256, 256, 0, stream>>>(out, h, deg, b, n);
}